// GATv2Layer_57964878626937
// MI455X (gfx1250) — compile-verified
//
#include <hip/hip_runtime.h>
#include <math.h>

typedef __attribute__((ext_vector_type(2))) float v2f;
typedef __attribute__((ext_vector_type(8))) float v8f;

#define VV   128   // vertices
#define DD   64    // attention dim
#define CO   256   // C_out
#define CI   256   // C_in
#define BB   256   // batch

// ---- async global->LDS staging (gfx1250), guarded so we always compile ----
#if defined(__has_builtin)
#if __has_builtin(__builtin_amdgcn_global_load_async_to_lds_b128)
#define HAVE_ASYNC128 1
#endif
#if __has_builtin(__builtin_amdgcn_global_load_async_to_lds_b32)
#define HAVE_ASYNC32 1
#endif
#endif

#if defined(HAVE_ASYNC128) || defined(HAVE_ASYNC32)
#define ASYNC_WAIT(N) asm volatile("s_wait_asynccnt %0" ::"n"(N) : "memory")
#else
#define ASYNC_WAIT(N)
#endif

#ifdef HAVE_ASYNC128
#define NB128 8
#else
#define NB128 0
#endif
#ifdef HAVE_ASYNC32
#define NB32 8
#else
#define NB32 0
#endif
#define K1_BATCH (NB128 + NB32)  // async instrs per wave per K-chunk (kernel 1)
#define K2_BATCH (2 * NB32)      // async instrs per wave per whT chunk (kernel 2)

__device__ __forceinline__ void stage16(const float* g, float* l) {
#ifdef HAVE_ASYNC128
  typedef int v4i __attribute__((vector_size(4 * sizeof(int))));
  typedef __attribute__((address_space(1))) v4i* gp_t;
  typedef __attribute__((address_space(3))) v4i* lp_t;
  __builtin_amdgcn_global_load_async_to_lds_b128((gp_t)g, (lp_t)l, 0, 0);
#else
  *(float4*)l = *(const float4*)g;
#endif
}

__device__ __forceinline__ void stage4(const float* g, float* l) {
#ifdef HAVE_ASYNC32
  typedef __attribute__((address_space(1))) int* gp_t;
  typedef __attribute__((address_space(3))) int* lp_t;
  __builtin_amdgcn_global_load_async_to_lds_b32((gp_t)g, (lp_t)l, 0, 0);
#else
  *l = *g;
#endif
}

// ---------------------------------------------------------------------------
// Kernel 1: C[M,N] = A[M,K] * W[K,N]  (f32, WMMA 16x16x4)
// Block tile 128(M) x 32(N); 8 waves own a 16x32 strip (2 accumulators).
// Double-buffered async LDS staging: chunk c+1 streams in while WMMA runs
// on chunk c. LDS row stride 68 dwords: 16B-aligned B128 destinations AND
// conflict-free b64 fragment reads (banks 4r,4r+1 per lane).
// ---------------------------------------------------------------------------
__global__ __launch_bounds__(256) void gemm_wmma_f32(
    const float* __restrict__ A, const float* __restrict__ W,
    float* __restrict__ C, int M, int N, int K) {
  __shared__ __align__(16) float Asl[2][128 * 68];
  __shared__ __align__(16) float Wsl[2][32 * 68];

  const int tid  = threadIdx.x;
  const int wave = tid >> 5;
  const int lane = tid & 31;
  const int lm   = lane & 15;
  const int hi   = lane >> 4;          // lane-half selects K offset 0 / 2
  const int m0   = blockIdx.x * 128;
  const int n0   = blockIdx.y * 32;

  v8f acc0 = {};
  v8f acc1 = {};

  auto issue = [&](int k0, int buf) {
    {  // A tile 128x64 (async B128)
      const int r  = tid >> 4;          // 0..15
      const int kk = (tid & 15) << 2;   // 0..60
      #pragma unroll
      for (int it = 0; it < 8; ++it)
        stage16(&A[(size_t)(m0 + r + 16 * it) * K + k0 + kk],
                &Asl[buf][(r + 16 * it) * 68 + kk]);
    }
    {  // W tile 64x32, transposed -> [n][k] (async B32)
      const int n  = tid & 31;
      const int kb = tid >> 5;          // 0..7
      #pragma unroll
      for (int it = 0; it < 8; ++it)
        stage4(&W[(size_t)(k0 + kb + 8 * it) * N + n0 + n],
               &Wsl[buf][n * 68 + kb + 8 * it]);
    }
  };

  const int nchunk = K >> 6;  // K=256 -> 4
  issue(0, 0);
  for (int c = 0; c < nchunk; ++c) {
    const int cur = c & 1;
    if (c + 1 < nchunk) {
      issue((c + 1) << 6, cur ^ 1);
      ASYNC_WAIT(K1_BATCH);   // chunk c complete, chunk c+1 in flight
    } else {
      ASYNC_WAIT(0);
    }
    __syncthreads();

    const float* As = Asl[cur];
    const float* Ws = Wsl[cur];
    const int row = wave * 16 + lm;
    #pragma unroll
    for (int kk = 0; kk < 64; kk += 4) {
      v2f af  = *(const v2f*)&As[row * 68 + kk + 2 * hi];
      v2f bf0 = *(const v2f*)&Ws[lm * 68 + kk + 2 * hi];
      v2f bf1 = *(const v2f*)&Ws[(lm + 16) * 68 + kk + 2 * hi];
      acc0 = __builtin_amdgcn_wmma_f32_16x16x4_f32(
          false, af, false, bf0, (short)0, acc0, false, false);
      acc1 = __builtin_amdgcn_wmma_f32_16x16x4_f32(
          false, af, false, bf1, (short)0, acc1, false, false);
    }
    __syncthreads();   // buffer `cur` free for chunk c+2
  }

  // D layout: VGPR r -> M = r + 8*hi, N = lm
  float* Cp = &C[(size_t)(m0 + wave * 16 + hi * 8) * N + n0 + lm];
  #pragma unroll
  for (int r = 0; r < 8; ++r) {
    Cp[(size_t)r * N]      = acc0[r];
    Cp[(size_t)r * N + 16] = acc1[r];
  }
}

// ---------------------------------------------------------------------------
// Kernel 2: one block per batch.
//   sl/sr GEMVs  : e = 0.2*(sl_i+sr_j) + 0.8*sum_d a_d*relu(el+er)
//   Phase 1      : relu-term (3 VALU/elem), adj mask -> logits in LDS
//   Phase 2      : row softmax via wave32 shuffles
//   Phase 3      : out = ELU(alpha @ Wh) via WMMA, double-buffered Wh tiles
// ---------------------------------------------------------------------------
__global__ __launch_bounds__(256) void gat_attn_out(
    const float* __restrict__ el, const float* __restrict__ er,
    const float* __restrict__ Wh, const int* __restrict__ adj,
    const float* __restrict__ a, float* __restrict__ out) {
  __shared__ __align__(16) float Esm[VV * 130];    // logits -> alpha (padded)
  __shared__ __align__(16) float elsm[VV * 68];    // el rows (stride 68)
  __shared__ __align__(16) float whT[2][32 * 130]; // Wh tiles transposed [n][k]
  __shared__ float slsm[VV];                       // sl_i = a . el[i,:]

  const int b    = blockIdx.x;
  const int tid  = threadIdx.x;
  const int wave = tid >> 5;
  const int lane = tid & 31;
  const int lm   = lane & 15;
  const int hi   = lane >> 4;

  const float* elb = el + (size_t)b * VV * DD;
  const float* erb = er + (size_t)b * VV * DD;
  const float* whb = Wh + (size_t)b * VV * CO;

  // stage el rows into LDS (async B128)
  {
    const int r  = tid >> 4;
    const int kk = (tid & 15) << 2;
    #pragma unroll
    for (int it = 0; it < 8; ++it)
      stage16(&elb[(size_t)(r + 16 * it) * DD + kk],
              &elsm[(r + 16 * it) * 68 + kk]);
  }

  // a[] via uniform loads (scalarizable), er[j,:] into per-lane registers
  float areg[DD];
  #pragma unroll
  for (int d = 0; d < DD; d += 4) {
    float4 v = *(const float4*)&a[d];
    areg[d] = v.x; areg[d + 1] = v.y; areg[d + 2] = v.z; areg[d + 3] = v.w;
  }
  const int j = (wave & 3) * 32 + lane;   // waves 0-3 / 4-7 both span j=0..127
  float ereg[DD];
  #pragma unroll
  for (int d = 0; d < DD; d += 4) {
    float4 v = *(const float4*)&erb[(size_t)j * DD + d];
    ereg[d] = v.x; ereg[d + 1] = v.y; ereg[d + 2] = v.z; ereg[d + 3] = v.w;
  }
  // sr_j = a . er[j,:]  (register-resident)
  float srj = 0.f;
  #pragma unroll
  for (int d = 0; d < DD; ++d) srj += areg[d] * ereg[d];

  ASYNC_WAIT(0);
  __syncthreads();

  // sl_i = a . el[i,:] : lane -> (row = w*16+lm, d-half = hi), shuffle-combine
  {
    const int i = wave * 16 + lm;
    float p = 0.f;
    if (hi == 0) {
      #pragma unroll
      for (int d = 0; d < 32; ++d) p += areg[d] * elsm[i * 68 + d];
    } else {
      #pragma unroll
      for (int d = 0; d < 32; ++d) p += areg[32 + d] * elsm[i * 68 + 32 + d];
    }
    p += __shfl_xor(p, 16, 32);
    if (lane < 16) slsm[i] = p;
  }
  __syncthreads();

  // ---- Phase 1: relu-term logits ----
  const int i0 = (wave >> 2) * 64;        // waves 0-3: i 0..63, 4-7: 64..127
  for (int ii = 0; ii < 64; ++ii) {
    const int i = i0 + ii;
    float acc = 0.f;
    #pragma unroll
    for (int d4 = 0; d4 < DD; d4 += 4) {
      float4 ev = *(const float4*)&elsm[i * 68 + d4];   // uniform -> broadcast
      float z;
      z = ev.x + ereg[d4 + 0]; acc += areg[d4 + 0] * fmaxf(z, 0.f);
      z = ev.y + ereg[d4 + 1]; acc += areg[d4 + 1] * fmaxf(z, 0.f);
      z = ev.z + ereg[d4 + 2]; acc += areg[d4 + 2] * fmaxf(z, 0.f);
      z = ev.w + ereg[d4 + 3]; acc += areg[d4 + 3] * fmaxf(z, 0.f);
    }
    const float e = fmaf(0.8f, acc, 0.2f * (slsm[i] + srj));
    const int ad = adj[i * VV + j];
    Esm[i * 130 + j] = ad ? e : -__builtin_huge_valf();
  }
  __syncthreads();

  // ---- Phase 2: softmax over j, 16 rows per wave ----
  for (int rr = 0; rr < 16; ++rr) {
    const int i = wave * 16 + rr;
    float v[4];
    float mx = -__builtin_huge_valf();
    #pragma unroll
    for (int q = 0; q < 4; ++q) {
      v[q] = Esm[i * 130 + lane + 32 * q];
      mx = fmaxf(mx, v[q]);
    }
    #pragma unroll
    for (int off = 16; off >= 1; off >>= 1)
      mx = fmaxf(mx, __shfl_xor(mx, off, 32));
    float s = 0.f;
    #pragma unroll
    for (int q = 0; q < 4; ++q) { v[q] = expf(v[q] - mx); s += v[q]; }
    #pragma unroll
    for (int off = 16; off >= 1; off >>= 1) s += __shfl_xor(s, off, 32);
    const float inv = 1.0f / s;
    #pragma unroll
    for (int q = 0; q < 4; ++q) Esm[i * 130 + lane + 32 * q] = v[q] * inv;
  }
  __syncthreads();

  // ---- Phase 3: out = ELU(alpha @ Wh), double-buffered 32-col Wh tiles ----
  auto issueWh = [&](int nb, int buf) {
    const int n  = tid & 31;
    const int kb = tid >> 5;          // 0..7
    #pragma unroll
    for (int it = 0; it < 16; ++it)
      stage4(&whb[(size_t)(kb + 8 * it) * CO + nb * 32 + n],
             &whT[buf][n * 130 + kb + 8 * it]);
  };

  issueWh(0, 0);
  for (int nb = 0; nb < CO / 32; ++nb) {
    const int cur = nb & 1;
    if (nb + 1 < CO / 32) {
      issueWh(nb + 1, cur ^ 1);
      ASYNC_WAIT(K2_BATCH);
    } else {
      ASYNC_WAIT(0);
    }
    __syncthreads();

    v8f acc0 = {};
    v8f acc1 = {};
    const float* Ws = whT[cur];
    const int row = wave * 16 + lm;
    #pragma unroll
    for (int kk = 0; kk < VV; kk += 4) {
      v2f af  = *(const v2f*)&Esm[row * 130 + kk + 2 * hi];
      v2f bf0 = *(const v2f*)&Ws[lm * 130 + kk + 2 * hi];
      v2f bf1 = *(const v2f*)&Ws[(lm + 16) * 130 + kk + 2 * hi];
      acc0 = __builtin_amdgcn_wmma_f32_16x16x4_f32(
          false, af, false, bf0, (short)0, acc0, false, false);
      acc1 = __builtin_amdgcn_wmma_f32_16x16x4_f32(
          false, af, false, bf1, (short)0, acc1, false, false);
    }

    float* op = out + (size_t)b * VV * CO +
                (size_t)(wave * 16 + hi * 8) * CO + nb * 32 + lm;
    #pragma unroll
    for (int r = 0; r < 8; ++r) {
      const float x0 = acc0[r];
      const float x1 = acc1[r];
      op[(size_t)r * CO]      = x0 > 0.f ? x0 : (expf(x0) - 1.0f);
      op[(size_t)r * CO + 16] = x1 > 0.f ? x1 : (expf(x1) - 1.0f);
    }
    __syncthreads();
  }
}

// ---------------------------------------------------------------------------
extern "C" void kernel_launch(void* const* d_in, const int* in_sizes, int n_in,
                              void* d_out, int out_size, void* d_ws, size_t ws_size,
                              hipStream_t stream) {
  (void)in_sizes; (void)n_in; (void)out_size; (void)ws_size;
  const float* x     = (const float*)d_in[0];
  const int*   adj   = (const int*)d_in[1];
  const float* W_l   = (const float*)d_in[2];
  const float* W_r   = (const float*)d_in[3];
  const float* a     = (const float*)d_in[4];
  const float* W_out = (const float*)d_in[5];
  float* out = (float*)d_out;

  const int M = BB * VV;   // 32768
  float* ws = (float*)d_ws;
  float* Wh = ws;                                  // M * 256
  float* el = Wh + (size_t)M * CO;                 // M * 64
  float* er = el + (size_t)M * DD;                 // M * 64

  dim3 blk(256);
  gemm_wmma_f32<<<dim3(M / 128, CO / 32), blk, 0, stream>>>(x, W_out, Wh, M, CO, CI);
  gemm_wmma_f32<<<dim3(M / 128, DD / 32), blk, 0, stream>>>(x, W_l,   el, M, DD, CI);
  gemm_wmma_f32<<<dim3(M / 128, DD / 32), blk, 0, stream>>>(x, W_r,   er, M, DD, CI);
  gat_attn_out<<<dim3(BB), blk, 0, stream>>>(el, er, Wh, adj, a, out);
}